// NeuralRendering_10342281249293
// MI455X (gfx1250) — compile-verified
//
#include <hip/hip_runtime.h>
#include <hip/hip_bf16.h>
#include <cstdint>

// ---------------------------------------------------------------------------
// NeRF volume-rendering composite for MI455X (gfx1250).
// Memory-bound streaming kernel (~390 MiB @ 23.3 TB/s => ~17us floor).
// One ray per lane; raw[] staged to LDS via async DMA (ASYNCcnt) for
// coalesced 128B global traffic, double-buffered to overlap DMA with the
// per-ray transmittance scan.
// ---------------------------------------------------------------------------

#define R_TOTAL 131072
#define S_SAMP  128
#define CHUNK   8            // samples per staged chunk
#define NCH     (S_SAMP / CHUNK)   // 16 chunks
#define RPB     128          // rays per block == threads per block
#define ROWD    36           // dwords per LDS row: 8 samples * 4 floats + 4 pad
#define BIG_DIST 1e10f
#define EPS      1e-10f

// ---- gfx1250 async global->LDS DMA (ASYNCcnt) -----------------------------
#if defined(__AMDGCN__) && defined(__has_builtin)
#  if __has_builtin(__builtin_amdgcn_global_load_async_to_lds_b128)
#    define HAVE_ASYNC_LDS 1
#  endif
#endif
#ifndef HAVE_ASYNC_LDS
#  define HAVE_ASYNC_LDS 0
#endif

#if defined(__AMDGCN__) && defined(__has_builtin) && __has_builtin(__builtin_amdgcn_s_wait_asynccnt)
#  define WAIT_ASYNC() __builtin_amdgcn_s_wait_asynccnt(0)
#elif HAVE_ASYNC_LDS
#  define WAIT_ASYNC() asm volatile("s_wait_asynccnt 0" ::: "memory")
#else
#  define WAIT_ASYNC() do {} while (0)
#endif

#if HAVE_ASYNC_LDS
typedef int v4i __attribute__((ext_vector_type(4)));
typedef __attribute__((address_space(1))) v4i g_v4i;   // global int4*
typedef __attribute__((address_space(3))) v4i l_v4i;   // LDS int4*
#endif

__device__ __forceinline__ void async_copy16(const float* __restrict__ g,
                                             float* __restrict__ l) {
#if HAVE_ASYNC_LDS
  // Global AS1 pointer: flat and global share numeric addresses.
  g_v4i* gp = (g_v4i*)(unsigned long long)(uintptr_t)g;
  // LDS AS3 pointer: low 32 bits of a flat LDS address are the LDS offset.
  l_v4i* lp = (l_v4i*)(unsigned int)(uintptr_t)l;
  __builtin_amdgcn_global_load_async_to_lds_b128(gp, lp, /*offset=*/0, /*cpol=*/0);
#else
  *(float4*)l = *(const float4*)g;
#endif
}

// Stage one chunk (8 samples) of raw[] for 128 rays into an LDS buffer.
// Thread t: 8 threads cover one ray's contiguous 128B chunk-row (fully
// coalesced 128B global segments); 8 iterations walk 16-ray groups.
__device__ __forceinline__ void stage_chunk(const float* __restrict__ raw,
                                            float* __restrict__ ldsbuf,
                                            int blockBase, int c, int tid) {
  const int grow = tid >> 3;   // 0..15: ray-group row
  const int gcol = tid & 7;    // 0..7 : 16B column within 128B row
  const float* g = raw + (size_t)(blockBase + grow) * (S_SAMP * 4)
                       + (size_t)c * (CHUNK * 4) + gcol * 4;
  float* l = ldsbuf + grow * ROWD + gcol * 4;
#pragma unroll
  for (int k = 0; k < 8; ++k) {
    async_copy16(g, l);
    g += (size_t)16 * (S_SAMP * 4);  // advance 16 rays in global
    l += 16 * ROWD;                  // advance 16 rows in LDS
  }
}

__global__ __launch_bounds__(RPB) void nerf_composite_kernel(
    const float* __restrict__ raw,   // [R, S, 4]
    const float* __restrict__ zv,    // [R, S]
    const float* __restrict__ rdv,   // [R, 3]
    float* __restrict__ out) {       // rgb(3R) | disp(R) | acc(R) | w(128R) | depth(R)
  __shared__ float sbuf[2][RPB * ROWD];   // 2 x 18KiB double buffer

  const int tid       = threadIdx.x;
  const int blockBase = blockIdx.x * RPB;
  const int ray       = blockBase + tid;

  // |rays_d| (rays_d is tiny; L2-resident)
  const float dx = rdv[(size_t)ray * 3 + 0];
  const float dy = rdv[(size_t)ray * 3 + 1];
  const float dz = rdv[(size_t)ray * 3 + 2];
  const float dn = sqrtf(dx * dx + dy * dy + dz * dz);

  const float* zray = zv + (size_t)ray * S_SAMP;
  float* wout = out + (size_t)5 * R_TOTAL + (size_t)ray * S_SAMP;

  // Prologue: stage chunk 0.
  stage_chunk(raw, &sbuf[0][0], blockBase, 0, tid);
  WAIT_ASYNC();
  __syncthreads();

  float T = 1.0f;                    // exclusive transmittance cumprod
  float accw = 0.0f, depth = 0.0f;
  float r_acc = 0.0f, g_acc = 0.0f, b_acc = 0.0f;

  for (int c = 0; c < NCH; ++c) {
    const int b = c & 1;
    // Prefetch next chunk into the other buffer (overlaps compute below).
    if (c + 1 < NCH) stage_chunk(raw, &sbuf[b ^ 1][0], blockBase, c + 1, tid);

    // z for this chunk + one-sample lookahead for the last interval.
    const float4 z0 = *(const float4*)(zray + c * CHUNK);
    const float4 z1 = *(const float4*)(zray + c * CHUNK + 4);
    const float zz[8] = {z0.x, z0.y, z0.z, z0.w, z1.x, z1.y, z1.z, z1.w};
    float d[8];
#pragma unroll
    for (int j = 0; j < 7; ++j) d[j] = zz[j + 1] - zz[j];
    if (c + 1 < NCH) {
      d[7] = zray[c * CHUNK + 8] - zz[7];
    } else {
      d[7] = BIG_DIST;               // sentinel last interval
    }

    float w8[8];
    const float* lrow = &sbuf[b][tid * ROWD];   // 16B-aligned (144B rows)
#pragma unroll
    for (int j = 0; j < 8; ++j) {
      const float4 rv = *(const float4*)(lrow + j * 4);  // ds_load_b128
      const float sigma = fmaxf(rv.w, 0.0f);
      const float alpha = 1.0f - __expf(-sigma * d[j] * dn);
      const float w = alpha * T;
      T = T * (1.0f - alpha + EPS);
      const float sr = 1.0f / (1.0f + __expf(-rv.x));
      const float sg = 1.0f / (1.0f + __expf(-rv.y));
      const float sb = 1.0f / (1.0f + __expf(-rv.z));
      r_acc += w * sr;
      g_acc += w * sg;
      b_acc += w * sb;
      depth += w * zz[j];
      accw  += w;
      w8[j] = w;
    }
    *(float4*)(wout + c * CHUNK)     = make_float4(w8[0], w8[1], w8[2], w8[3]);
    *(float4*)(wout + c * CHUNK + 4) = make_float4(w8[4], w8[5], w8[6], w8[7]);

    if (c + 1 < NCH) {               // uniform branch: safe with barrier
      WAIT_ASYNC();                  // our prefetch for buf[b^1] landed
      __syncthreads();               // everyone's prefetch landed / reads done
    }
  }

  // Final per-ray outputs (coalesced scalar stores).
  out[(size_t)ray * 3 + 0] = r_acc;
  out[(size_t)ray * 3 + 1] = g_acc;
  out[(size_t)ray * 3 + 2] = b_acc;
  const float dmap = depth / accw;
  out[(size_t)3 * R_TOTAL + ray]   = 1.0f / fmaxf(EPS, dmap);  // disp
  out[(size_t)4 * R_TOTAL + ray]   = accw;                     // acc
  out[(size_t)133 * R_TOTAL + ray] = depth;                    // depth
}

extern "C" void kernel_launch(void* const* d_in, const int* in_sizes, int n_in,
                              void* d_out, int out_size, void* d_ws, size_t ws_size,
                              hipStream_t stream) {
  (void)in_sizes; (void)n_in; (void)out_size; (void)d_ws; (void)ws_size;
  const float* raw = (const float*)d_in[0];   // [131072,128,4] f32
  const float* zv  = (const float*)d_in[1];   // [131072,128]   f32
  const float* rdv = (const float*)d_in[2];   // [131072,3]     f32
  float* out = (float*)d_out;                 // 134*R floats

  dim3 grid(R_TOTAL / RPB);   // 1024 blocks
  dim3 block(RPB);            // 128 threads = 4 wave32
  nerf_composite_kernel<<<grid, block, 0, stream>>>(raw, zv, rdv, out);
}